// CustomGATLayer_53309134078172
// MI455X (gfx1250) — compile-verified
//
#include <hip/hip_runtime.h>

typedef __attribute__((ext_vector_type(2))) float v2f;
typedef __attribute__((ext_vector_type(8))) float v8f;

#define GN   50000
#define GE   800000
#define GDIN 128
#define GDOUT 128
#define GH   4
#define GHD  32

// ---------------------------------------------------------------------------
// Kernel 0: workspace init. grid = (N*128)/256 = 25000
// ---------------------------------------------------------------------------
__global__ __launch_bounds__(256) void init_kernel(float* __restrict__ vsum,
                                                   float* __restrict__ denom,
                                                   float* __restrict__ wsum,
                                                   unsigned* __restrict__ maxkey,
                                                   float* __restrict__ deg) {
    int i = blockIdx.x * 256 + threadIdx.x;   // i < N*128 exactly
    vsum[i] = 0.0f;
    if (i < GN * GH) {
        denom[i]  = 0.0f;
        wsum[i]   = 0.0f;
        maxkey[i] = 0x007FFFFFu;              // monotone-encoded -inf
    }
    if (i < GN) deg[i] = 0.0f;
}

// ---------------------------------------------------------------------------
// Kernel 1: fused Q/K/V GEMM with fp32 WMMA (V_WMMA_F32_16X16X4_F32).
// 32 rows per block, 8 waves. Wave w owns column strips 3w..3w+2 (of 24 =
// 3 matrices x 8 column tiles) for BOTH 16-row tiles: each W strip is read
// by exactly one wave (W L2 traffic halved vs 16-row blocks) and each B
// fragment feeds two WMMAs. 6 x v8f accumulators per wave.
// Tail block (rows >= 50000): A-load rows clamped (EXEC stays all-ones for
// WMMA), stores predicated.
// ---------------------------------------------------------------------------
__global__ __launch_bounds__(256) void qkv_wmma_kernel(
    const float* __restrict__ x,
    const float* __restrict__ Wq, const float* __restrict__ bq,
    const float* __restrict__ Wk, const float* __restrict__ bk,
    const float* __restrict__ Wv, const float* __restrict__ bv,
    float* __restrict__ qo, float* __restrict__ ko, float* __restrict__ vo)
{
    const int tid  = threadIdx.x;
    const int lane = tid & 31;
    const int wave = __builtin_amdgcn_readfirstlane(tid >> 5);   // 0..7, scalar
    const int row0 = blockIdx.x * 32;
    const int m    = lane & 15;
    const int hi   = lane >> 4;            // lane half: K offset 0 or 2
    const int ct0  = wave * 3;             // first of this wave's 3 col strips

    const float* Wmat[3] = {Wq, Wk, Wv};
    float*       Omat[3] = {qo, ko, vo};
    const float* Bias[3] = {bq, bk, bv};

    v8f acc[2][3];
    #pragma unroll
    for (int rt = 0; rt < 2; ++rt)
        #pragma unroll
        for (int t = 0; t < 3; ++t) acc[rt][t] = (v8f){};

    // A row pointers for the two row tiles (clamped for the tail block)
    int ar0 = row0 + m;       if (ar0 > GN - 1) ar0 = GN - 1;
    int ar1 = row0 + 16 + m;  if (ar1 > GN - 1) ar1 = GN - 1;
    const float* xrow0 = x + (size_t)ar0 * GDIN;
    const float* xrow1 = x + (size_t)ar1 * GDIN;

    // B row pointers / bias per column strip (mat index is scalar)
    const float* Wrow[3];
    float biasv[3];
    int   ncol[3];
    #pragma unroll
    for (int t = 0; t < 3; ++t) {
        const int ct  = ct0 + t;           // 0..23, scalar
        const int mat = ct >> 3;           // scalar -> s_cselect of base ptr
        const int n   = ((ct & 7) << 4) + m;
        ncol[t]  = n;
        Wrow[t]  = Wmat[mat] + (size_t)n * GDIN;
        biasv[t] = Bias[mat][n];
    }

    #pragma unroll 4
    for (int kk = 0; kk < GDIN; kk += 4) {
        const int kb = kk + (hi << 1);
        v2f a0, a1;
        a0.x = xrow0[kb]; a0.y = xrow0[kb + 1];
        a1.x = xrow1[kb]; a1.y = xrow1[kb + 1];
        #pragma unroll
        for (int t = 0; t < 3; ++t) {
            v2f b;
            b.x = Wrow[t][kb]; b.y = Wrow[t][kb + 1];
            acc[0][t] = __builtin_amdgcn_wmma_f32_16x16x4_f32(
                false, a0, false, b, (short)0, acc[0][t], false, false);
            acc[1][t] = __builtin_amdgcn_wmma_f32_16x16x4_f32(
                false, a1, false, b, (short)0, acc[1][t], false, false);
        }
    }

    #pragma unroll
    for (int rt = 0; rt < 2; ++rt) {
        const int mbase = row0 + rt * 16 + (hi << 3);
        #pragma unroll
        for (int t = 0; t < 3; ++t) {
            const int ct  = ct0 + t;
            const int mat = ct >> 3;
            float* outc = Omat[mat] + ncol[t];
            #pragma unroll
            for (int r = 0; r < 8; ++r) {
                const int mrow = mbase + r;
                if (mrow < GN)
                    outc[(size_t)mrow * GDOUT] = acc[rt][t][r] + biasv[t];
            }
        }
    }
}

// ---------------------------------------------------------------------------
// Kernel 2: per-(node, head) attention dot products sq = q.a_q, sk = k.a_k
// ---------------------------------------------------------------------------
__global__ __launch_bounds__(256) void attdot_kernel(
    const float* __restrict__ q, const float* __restrict__ k,
    const float* __restrict__ att,
    float* __restrict__ sq, float* __restrict__ sk)
{
    int idx = blockIdx.x * 256 + threadIdx.x;
    if (idx >= GN * GH) return;
    int n = idx >> 2, h = idx & 3;
    const float4* qp  = (const float4*)(q   + (size_t)n * GDOUT + h * GHD);
    const float4* kp  = (const float4*)(k   + (size_t)n * GDOUT + h * GHD);
    const float4* aqp = (const float4*)(att + h * 2 * GHD);
    const float4* akp = (const float4*)(att + h * 2 * GHD + GHD);
    float s1 = 0.f, s2 = 0.f;
    #pragma unroll
    for (int j = 0; j < GHD / 4; ++j) {
        float4 a = qp[j], b = aqp[j];
        s1 += a.x * b.x + a.y * b.y + a.z * b.z + a.w * b.w;
        float4 c = kp[j], d = akp[j];
        s2 += c.x * d.x + c.y * d.y + c.z * d.z + c.w * d.w;
    }
    sq[idx] = s1;
    sk[idx] = s2;
}

__device__ __forceinline__ unsigned enc_ord(float f) {
    unsigned u = __float_as_uint(f);
    return (u & 0x80000000u) ? ~u : (u | 0x80000000u);
}
__device__ __forceinline__ float dec_ord(unsigned u) {
    return (u & 0x80000000u) ? __uint_as_float(u ^ 0x80000000u)
                             : __uint_as_float(~u);
}

// ---------------------------------------------------------------------------
// Kernel 3: per-edge scores + segment max (ordered-uint atomicMax) + degree.
// grid = E/256 = 3125
// ---------------------------------------------------------------------------
__global__ __launch_bounds__(256) void edge_scores_kernel(
    const long long* __restrict__ ei,
    const float* __restrict__ sq, const float* __restrict__ sk,
    float* __restrict__ scores, unsigned* __restrict__ maxkey,
    float* __restrict__ deg)
{
    int e = blockIdx.x * 256 + threadIdx.x;
    int row = (int)ei[e];
    int col = (int)ei[GE + e];
    #pragma unroll
    for (int h = 0; h < GH; ++h) {
        float s = sq[row * GH + h] + sk[col * GH + h];
        s = (s > 0.f) ? s : 0.2f * s;          // leaky_relu(0.2)
        scores[(size_t)e * GH + h] = s;
        atomicMax(&maxkey[col * GH + h], enc_ord(s));
    }
    atomicAdd(&deg[col], 1.0f);
}

// ---------------------------------------------------------------------------
// Kernel 4: e = exp(score - m[col]); denom += e (scores overwritten with e).
// grid = E*4/256 = 12500
// ---------------------------------------------------------------------------
__global__ __launch_bounds__(256) void edge_exp_kernel(
    const long long* __restrict__ ei,
    float* __restrict__ scores, const unsigned* __restrict__ maxkey,
    float* __restrict__ denom)
{
    int idx = blockIdx.x * 256 + threadIdx.x;   // < E*4
    int e = idx >> 2, h = idx & 3;
    int col = (int)ei[GE + e];
    float mx = dec_ord(maxkey[col * GH + h]);
    float ev = expf(scores[idx] - mx);
    scores[idx] = ev;
    atomicAdd(&denom[col * GH + h], ev);
}

// ---------------------------------------------------------------------------
// Kernel 5: wsum += e / denom[col].  grid = 12500
// ---------------------------------------------------------------------------
__global__ __launch_bounds__(256) void edge_wsum_kernel(
    const long long* __restrict__ ei,
    const float* __restrict__ scores, const float* __restrict__ denom,
    float* __restrict__ wsum)
{
    int idx = blockIdx.x * 256 + threadIdx.x;
    int e = idx >> 2, h = idx & 3;
    int col = (int)ei[GE + e];
    atomicAdd(&wsum[col * GH + h], scores[idx] / denom[col * GH + h]);
}

// ---------------------------------------------------------------------------
// Kernel 6: segment-sum of v[row] into vsum[col]: one float4 per task.
// grid = E*32/256 = 100000
// ---------------------------------------------------------------------------
__global__ __launch_bounds__(256) void scatter_v_kernel(
    const long long* __restrict__ ei,
    const float* __restrict__ v, float* __restrict__ vsum)
{
    int t = blockIdx.x * 256 + threadIdx.x;     // < E*32 (25.6M < 2^31)
    int e = t >> 5, j = t & 31;
    int row = (int)ei[e];
    int col = (int)ei[GE + e];
    float4 val = ((const float4*)(v + (size_t)row * GDOUT))[j];
    float* dst = vsum + (size_t)col * GDOUT + (j << 2);
    atomicAdd(dst + 0, val.x);
    atomicAdd(dst + 1, val.y);
    atomicAdd(dst + 2, val.z);
    atomicAdd(dst + 3, val.w);
}

// ---------------------------------------------------------------------------
// Kernel 7: out = vsum * attn_w, attn_w = deg>0 ? wsum/max(deg,1) : 0.
// grid = 25000
// ---------------------------------------------------------------------------
__global__ __launch_bounds__(256) void finalize_kernel(
    const float* __restrict__ vsum, const float* __restrict__ wsum,
    const float* __restrict__ deg, float* __restrict__ out)
{
    int i = blockIdx.x * 256 + threadIdx.x;     // < N*128
    int n = i >> 7;
    int h = (i & 127) >> 5;
    float dg = deg[n];
    float aw = (dg > 0.f) ? wsum[n * GH + h] / fmaxf(dg, 1.f) : 0.f;
    out[i] = vsum[i] * aw;
}

// ---------------------------------------------------------------------------
extern "C" void kernel_launch(void* const* d_in, const int* in_sizes, int n_in,
                              void* d_out, int out_size, void* d_ws, size_t ws_size,
                              hipStream_t stream) {
    const float*     x   = (const float*)d_in[0];
    const long long* ei  = (const long long*)d_in[1];   // int64 (2, E)
    const float*     Wq  = (const float*)d_in[2];
    const float*     bq  = (const float*)d_in[3];
    const float*     Wk  = (const float*)d_in[4];
    const float*     bk  = (const float*)d_in[5];
    const float*     Wv  = (const float*)d_in[6];
    const float*     bv  = (const float*)d_in[7];
    const float*     att = (const float*)d_in[8];
    float* out = (float*)d_out;

    // workspace carve-up (~120 MB)
    float* q      = (float*)d_ws;                 // N*128
    float* k      = q    + (size_t)GN * GDOUT;    // N*128
    float* v      = k    + (size_t)GN * GDOUT;    // N*128
    float* vsum   = v    + (size_t)GN * GDOUT;    // N*128
    float* sq     = vsum + (size_t)GN * GDOUT;    // N*4
    float* sk     = sq   + (size_t)GN * GH;       // N*4
    float* denom  = sk   + (size_t)GN * GH;       // N*4
    float* wsum   = denom+ (size_t)GN * GH;       // N*4
    unsigned* mk  = (unsigned*)(wsum + (size_t)GN * GH); // N*4
    float* deg    = (float*)(mk + (size_t)GN * GH);      // N
    float* scores = deg + GN;                     // E*4

    init_kernel<<<(GN * GDOUT) / 256, 256, 0, stream>>>(vsum, denom, wsum, mk, deg);
    qkv_wmma_kernel<<<(GN + 31) / 32, 256, 0, stream>>>(x, Wq, bq, Wk, bk, Wv, bv, q, k, v);
    attdot_kernel<<<(GN * GH + 255) / 256, 256, 0, stream>>>(q, k, att, sq, sk);
    edge_scores_kernel<<<GE / 256, 256, 0, stream>>>(ei, sq, sk, scores, mk, deg);
    edge_exp_kernel<<<(GE * GH) / 256, 256, 0, stream>>>(ei, scores, mk, denom);
    edge_wsum_kernel<<<(GE * GH) / 256, 256, 0, stream>>>(ei, scores, denom, wsum);
    scatter_v_kernel<<<(GE * 32) / 256, 256, 0, stream>>>(ei, v, vsum);
    finalize_kernel<<<(GN * GDOUT) / 256, 256, 0, stream>>>(vsum, wsum, deg, out);
}